// GNNLayer_62285615726744
// MI455X (gfx1250) — compile-verified
//
#include <hip/hip_runtime.h>

typedef __attribute__((ext_vector_type(2))) float v2f;
typedef __attribute__((ext_vector_type(8))) float v8f;
typedef __attribute__((ext_vector_type(4))) int   v4i;

#define D_FEAT 128

#define AS1 __attribute__((address_space(1)))
#define AS3 __attribute__((address_space(3)))

#if defined(__has_builtin)
#  if __has_builtin(__builtin_amdgcn_global_load_async_to_lds_b128)
#    define HAS_ASYNC_LDS 1
#  else
#    define HAS_ASYNC_LDS 0
#  endif
#else
#  define HAS_ASYNC_LDS 0
#endif

// ---------------------------------------------------------------------------
// Kernel 1: zero the segment-sum accumulator (d_ws) every call (determinism).
// ---------------------------------------------------------------------------
__global__ void gnn_zero_ws(float4* __restrict__ h4, long long n4) {
    long long i = (long long)blockIdx.x * blockDim.x + threadIdx.x;
    long long stride = (long long)gridDim.x * blockDim.x;
    float4 z = {0.f, 0.f, 0.f, 0.f};
    for (; i < n4; i += stride) h4[i] = z;
}

// ---------------------------------------------------------------------------
// Kernel 2: edge scatter-add. One wave (32 lanes) per edge; each lane moves a
// float4 (32 lanes * 4 = 128 features) and issues 4 native f32 atomics
// (global_atomic_add_f32). x and h both fit in the 192MB L2, so these
// atomics resolve in L2, not HBM.
// ---------------------------------------------------------------------------
__global__ void gnn_scatter_add(const float* __restrict__ x,
                                const int* __restrict__ src,
                                const int* __restrict__ dst,
                                float* __restrict__ h, int E) {
    int wave = blockIdx.x * (blockDim.x >> 5) + (threadIdx.x >> 5);
    int lane = threadIdx.x & 31;
    if (wave >= E) return;
    int s = src[wave];
    int d = dst[wave];
    const float4 v = ((const float4*)(x + (size_t)s * D_FEAT))[lane];
    float* hp = h + (size_t)d * D_FEAT + lane * 4;
    unsafeAtomicAdd(hp + 0, v.x);
    unsafeAtomicAdd(hp + 1, v.y);
    unsafeAtomicAdd(hp + 2, v.z);
    unsafeAtomicAdd(hp + 3, v.w);
}

// ---------------------------------------------------------------------------
// Kernel 3: out[N,128] = h[N,128] @ W^T + b  via V_WMMA_F32_16X16X4_F32.
//
// One block = one 16-row tile of h; its 8 waves cover the 8 column tiles.
// The 16x128 h tile (8KB) is staged into LDS ONCE per block via the CDNA5
// async global->LDS DMA path (GLOBAL_LOAD_ASYNC_TO_LDS_B128, ASYNCcnt),
// eliminating the 8x redundant A-fragment global reads; A fragments then
// come from LDS (ds_load_b64). W (64KB) stays on the global path — it is
// L2-resident and shared by every block.
//
// Fragment layouts (ISA 7.12.2, 32-bit, wave32):
//   A 16x4 : lane<16 -> M=lane, v0=K0, v1=K1 ; lane>=16 -> M=lane-16, K2/K3
//   B 4x16 : lane<16 -> N=lane, v0=K0 row, v1=K1 ; lane>=16 -> N=lane-16, K2/K3
//   C 16x16: vgpr i -> M = i + 8*(lane>=16), N = lane&15
// N_NODES = 100000 is divisible by 16, so EXEC is all-ones for every WMMA.
// ---------------------------------------------------------------------------
__global__ void gnn_gemm_wmma(const float* __restrict__ h,
                              const float* __restrict__ W,
                              const float* __restrict__ bias,
                              float* __restrict__ out,
                              int nRowTiles) {
    __shared__ __align__(16) float hs[16 * D_FEAT];   // 8 KB h tile

    const int rowTile = blockIdx.x;
    if (rowTile >= nRowTiles) return;                 // block-uniform
    const int rowBase = rowTile * 16;

    // ---- stage the 16x128 h tile into LDS ----
    // 512 float4 chunks, 256 threads -> 2 chunks each.
    const int tid = threadIdx.x;
    for (int c = tid; c < 512; c += 256) {
        const int row  = c >> 5;                      // 0..15
        const int col4 = (c & 31) << 2;               // 0,4,...,124
        const float* gp = h + (size_t)(rowBase + row) * D_FEAT + col4;
        float* lp = &hs[row * D_FEAT + col4];
#if HAS_ASYNC_LDS
        // builtin signature (from hipcc diagnostic): arg0 is non-const v4i*
        // in the global (__device__) address space.
        __builtin_amdgcn_global_load_async_to_lds_b128(
            (AS1 v4i*)(v4i*)gp,
            (AS3 v4i*)(v4i*)lp, /*offset=*/0, /*cpol=*/0);
#else
        *(float4*)lp = *(const float4*)gp;            // fallback: load + ds_store
#endif
    }
#if HAS_ASYNC_LDS
#  if defined(__has_builtin) && __has_builtin(__builtin_amdgcn_s_wait_asynccnt)
    __builtin_amdgcn_s_wait_asynccnt(0);
#  else
    asm volatile("s_wait_asynccnt 0x0" ::: "memory");
#  endif
#endif
    __syncthreads();

    // ---- WMMA: each wave owns one 16-column tile ----
    const int lane    = threadIdx.x & 31;
    const int halfSel = lane >> 4;                    // 0 / 1
    const int laneLo  = lane & 15;
    const int colBase = (threadIdx.x >> 5) * 16;      // wave id * 16

    const float* arow = &hs[laneLo * D_FEAT + 2 * halfSel];             // LDS
    const float* wrow = W + (size_t)(colBase + laneLo) * D_FEAT + 2 * halfSel;

    v8f c = {};
#pragma unroll
    for (int k0 = 0; k0 < D_FEAT; k0 += 4) {
        v2f a, bf;
        a[0]  = arow[k0];
        a[1]  = arow[k0 + 1];
        bf[0] = wrow[k0];
        bf[1] = wrow[k0 + 1];
        // 8 args: (neg_a, A, neg_b, B, c_mod, C, reuse_a, reuse_b)
        c = __builtin_amdgcn_wmma_f32_16x16x4_f32(
            false, a, false, bf, (short)0, c, false, false);
    }

    const float bv = bias[colBase + laneLo];
#pragma unroll
    for (int i = 0; i < 8; ++i) {
        const int m = rowBase + halfSel * 8 + i;
        out[(size_t)m * D_FEAT + colBase + laneLo] = c[i] + bv;
    }
}

// ---------------------------------------------------------------------------
extern "C" void kernel_launch(void* const* d_in, const int* in_sizes, int n_in,
                              void* d_out, int out_size, void* d_ws, size_t ws_size,
                              hipStream_t stream) {
    const float* x   = (const float*)d_in[0];
    const int*   src = (const int*)d_in[1];
    const int*   dst = (const int*)d_in[2];
    const float* W   = (const float*)d_in[3];
    const float* b   = (const float*)d_in[4];
    float* out = (float*)d_out;
    float* h   = (float*)d_ws;   // [N, 128] fp32 accumulator, 51.2 MB

    const int N = in_sizes[0] / D_FEAT;   // 100000
    const int E = in_sizes[1];            // 625000

    // 1) zero accumulator
    const long long n4 = (long long)N * D_FEAT / 4;
    gnn_zero_ws<<<2048, 256, 0, stream>>>((float4*)h, n4);

    // 2) scatter-add: one wave per edge, 8 waves per 256-thread block
    const int sblocks = (E + 7) / 8;
    gnn_scatter_add<<<sblocks, 256, 0, stream>>>(x, src, dst, h, E);

    // 3) WMMA GEMM + bias: one block per 16-row tile (LDS-staged), 8 waves
    //    cover the 8 column tiles. N divisible by 16 -> full EXEC everywhere.
    const int nRowTiles = N / 16;         // 6250
    gnn_gemm_wmma<<<nRowTiles, 256, 0, stream>>>(h, W, b, out, nRowTiles);
}